// DimeModule_21191368639069
// MI455X (gfx1250) — compile-verified
//
#include <hip/hip_runtime.h>
#include <hip/hip_bf16.h>

typedef float v2f __attribute__((ext_vector_type(2)));
typedef float v8f __attribute__((ext_vector_type(8)));

#define N_ATOMS 40000
#define N_EDGES 400000
#define N_MSG   1200000
#define DM 128
#define DR 64

__device__ __forceinline__ float swishf(float v) {
    return v / (1.0f + __expf(-v));
}

// D = A(16x4,f32) x B(4x16,f32) + C(16x16,f32)
__device__ __forceinline__ v8f wmma4(v2f a, v2f b, v8f c) {
    return __builtin_amdgcn_wmma_f32_16x16x4_f32(false, a, false, b, (short)0, c,
                                                 false, false);
}

// A fragment from row-major [rows x ld] global memory, tile rows [row0, row0+16)
__device__ __forceinline__ v2f ldAg(const float* __restrict__ P, int row0, int ld,
                                    int k0, int lane) {
    int r = row0 + (lane & 15);
    int k = k0 + ((lane >> 4) << 1);
    const float* p = P + (size_t)r * ld + k;
    v2f a; a.x = p[0]; a.y = p[1];
    return a;
}

// B fragment from row-major [K x 128] weight matrix in global memory
__device__ __forceinline__ v2f ldBg(const float* __restrict__ W, int k0, int n,
                                    int lane) {
    int k = k0 + ((lane >> 4) << 1);
    int c = (n << 4) + (lane & 15);
    v2f b; b.x = W[k * 128 + c]; b.y = W[(k + 1) * 128 + c];
    return b;
}

__global__ void k_zero(float* __restrict__ p, size_t n) {
    size_t i = (size_t)blockIdx.x * blockDim.x + threadIdx.x;
    size_t stride = (size_t)gridDim.x * blockDim.x;
    for (; i < n; i += stride) p[i] = 0.0f;
}

// x = swish(msg @ Wm + bm) * (rbf @ Wr)
__global__ __launch_bounds__(256, 1)
void k_edge_x(const float* __restrict__ msg, const float* __restrict__ rbf,
              const float* __restrict__ Wm, const float* __restrict__ bm,
              const float* __restrict__ Wr, float* __restrict__ xout) {
    int wave = threadIdx.x >> 5, lane = threadIdx.x & 31;
    int half = lane >> 4, l15 = lane & 15;
    int base = (blockIdx.x * 8 + wave) * 16;

    // rbf @ Wr first (16 k-steps), kept live through the second GEMM
    v8f accr[8] = {};
    for (int kk = 0; kk < 16; kk++) {
        v2f a = ldAg(rbf, base, DR, kk * 4, lane);
#pragma unroll
        for (int n = 0; n < 8; n++) accr[n] = wmma4(a, ldBg(Wr, kk * 4, n, lane), accr[n]);
    }
    // msg @ Wm (32 k-steps)
    v8f accm[8] = {};
    for (int kk = 0; kk < 32; kk++) {
        v2f a = ldAg(msg, base, DM, kk * 4, lane);
#pragma unroll
        for (int n = 0; n < 8; n++) accm[n] = wmma4(a, ldBg(Wm, kk * 4, n, lane), accm[n]);
    }
    // fused epilogue: x = swish(accm + b) * accr
#pragma unroll
    for (int n = 0; n < 8; n++) {
        float bb = bm[(n << 4) + l15];
#pragma unroll
        for (int v = 0; v < 8; v++) {
            size_t idx = (size_t)(base + v + 8 * half) * DM + (n << 4) + l15;
            xout[idx] = swishf(accm[n][v] + bb) * accr[n][v];
        }
    }
}

// m[w,:] = sum_l (x[src[w],:] * sbf8[w,l]) @ Wb[:,l,:]; atomic scatter by dst
__global__ __launch_bounds__(256, 1)
void k_bilinear(const float* __restrict__ x, const float* __restrict__ sbf,
                const float* __restrict__ Wsbf, const float* __restrict__ Wb,
                const int* __restrict__ eidx, float* __restrict__ accb) {
    extern __shared__ float sm[];
    float* sX  = sm;            // 8 waves * 16 rows * 128 = 16384 floats
    float* sWb = sm + 16384;    // 128 x 128 slice            16384 floats
    float* sWs = sm + 32768;    // 42 x 8                       336 floats
    int wave = threadIdx.x >> 5, lane = threadIdx.x & 31;
    int half = lane >> 4, l15 = lane & 15;
    int base = (blockIdx.x * 8 + wave) * 16;

    for (int i = threadIdx.x; i < 336; i += 256) sWs[i] = Wsbf[i];

    int myrow = base + l15;
    int src = eidx[myrow];
    {
        const float4* gp = (const float4*)(x + (size_t)src * DM + half * 64);
        float4* lp = (float4*)(sX + wave * 2048 + l15 * 128 + half * 64);
#pragma unroll
        for (int i = 0; i < 16; i++) lp[i] = gp[i];
    }
    __syncthreads();

    float s8[8];
#pragma unroll
    for (int o = 0; o < 8; o++) s8[o] = 0.0f;
    const float* sp = sbf + (size_t)myrow * 42;
    for (int j = 0; j < 42; j++) {
        float sv = sp[j];
#pragma unroll
        for (int o = 0; o < 8; o++) s8[o] += sv * sWs[j * 8 + o];
    }

    v8f acc[8] = {};
    for (int l8 = 0; l8 < 8; l8++) {
        __syncthreads();
        for (int t = threadIdx.x; t < 16384; t += 256) {
            int j = t >> 7, i = t & 127;
            sWb[t] = Wb[j * 1024 + l8 * 128 + i];
        }
        __syncthreads();
        float sc = s8[l8];
        for (int kk = 0; kk < 32; kk++) {
            int k = kk * 4 + half * 2;
            const float* ap = sX + wave * 2048 + l15 * 128 + k;
            v2f a; a.x = ap[0] * sc; a.y = ap[1] * sc;
#pragma unroll
            for (int n = 0; n < 8; n++) {
                int c = (n << 4) + l15;
                v2f b; b.x = sWb[k * 128 + c]; b.y = sWb[(k + 1) * 128 + c];
                acc[n] = wmma4(a, b, acc[n]);
            }
        }
    }

    int d[8];
#pragma unroll
    for (int v = 0; v < 8; v++) d[v] = eidx[N_MSG + base + v + 8 * half];
#pragma unroll
    for (int n = 0; n < 8; n++)
#pragma unroll
        for (int v = 0; v < 8; v++)
            atomicAdd(&accb[(size_t)d[v] * DM + (n << 4) + l15], acc[n][v]);
}

// out = in + swish(swish(in) @ W1 + b1) @ W2 + b2    (in-place safe per 16-row tile)
__global__ __launch_bounds__(256, 1)
void k_resblock(const float* __restrict__ in, float* __restrict__ out,
                const float* __restrict__ W1, const float* __restrict__ b1,
                const float* __restrict__ W2, const float* __restrict__ b2) {
    __shared__ float sG[8 * 2048];
    int wave = threadIdx.x >> 5, lane = threadIdx.x & 31;
    int half = lane >> 4, l15 = lane & 15;
    int base = (blockIdx.x * 8 + wave) * 16;

    v8f acc[8] = {};
    for (int kk = 0; kk < 32; kk++) {
        int k = kk * 4 + half * 2;
        const float* p = in + (size_t)(base + l15) * DM + k;
        v2f a; a.x = swishf(p[0]); a.y = swishf(p[1]);
#pragma unroll
        for (int n = 0; n < 8; n++) acc[n] = wmma4(a, ldBg(W1, kk * 4, n, lane), acc[n]);
    }
#pragma unroll
    for (int n = 0; n < 8; n++) {
        float bb = b1[(n << 4) + l15];
#pragma unroll
        for (int v = 0; v < 8; v++)
            sG[wave * 2048 + (v + 8 * half) * 128 + (n << 4) + l15] =
                swishf(acc[n][v] + bb);
    }
    v8f acc2[8] = {};
    for (int kk = 0; kk < 32; kk++) {
        int k = kk * 4 + half * 2;
        const float* ap = sG + wave * 2048 + l15 * 128 + k;
        v2f a; a.x = ap[0]; a.y = ap[1];
#pragma unroll
        for (int n = 0; n < 8; n++) acc2[n] = wmma4(a, ldBg(W2, kk * 4, n, lane), acc2[n]);
    }
#pragma unroll
    for (int n = 0; n < 8; n++) {
        float bb = b2[(n << 4) + l15];
#pragma unroll
        for (int v = 0; v < 8; v++) {
            size_t idx = (size_t)(base + v + 8 * half) * DM + (n << 4) + l15;
            out[idx] = in[idx] + acc2[n][v] + bb;
        }
    }
}

// out = swish(in @ W + b) + gate * msg
__global__ __launch_bounds__(256, 1)
void k_inter(const float* __restrict__ in, const float* __restrict__ msg,
             const float* __restrict__ gate, const float* __restrict__ W,
             const float* __restrict__ b, float* __restrict__ out) {
    int wave = threadIdx.x >> 5, lane = threadIdx.x & 31;
    int half = lane >> 4, l15 = lane & 15;
    int base = (blockIdx.x * 8 + wave) * 16;

    v8f acc[8] = {};
    for (int kk = 0; kk < 32; kk++) {
        v2f a = ldAg(in, base, DM, kk * 4, lane);
#pragma unroll
        for (int n = 0; n < 8; n++) acc[n] = wmma4(a, ldBg(W, kk * 4, n, lane), acc[n]);
    }
#pragma unroll
    for (int n = 0; n < 8; n++) {
        float bb = b[(n << 4) + l15];
        float gg = gate[(n << 4) + l15];
#pragma unroll
        for (int v = 0; v < 8; v++) {
            size_t idx = (size_t)(base + v + 8 * half) * DM + (n << 4) + l15;
            out[idx] = swishf(acc[n][v] + bb) + gg * msg[idx];
        }
    }
}

// a = (rbf @ Wor) * mji, atomic scatter-add to atoms
__global__ __launch_bounds__(256, 1)
void k_outedge(const float* __restrict__ rbf, const float* __restrict__ Wor,
               const float* __restrict__ mji, const int* __restrict__ aeidx,
               float* __restrict__ atomacc) {
    int wave = threadIdx.x >> 5, lane = threadIdx.x & 31;
    int half = lane >> 4, l15 = lane & 15;
    int base = (blockIdx.x * 8 + wave) * 16;

    v8f acc[8] = {};
    for (int kk = 0; kk < 16; kk++) {
        v2f a = ldAg(rbf, base, DR, kk * 4, lane);
#pragma unroll
        for (int n = 0; n < 8; n++) acc[n] = wmma4(a, ldBg(Wor, kk * 4, n, lane), acc[n]);
    }
    int d[8];
#pragma unroll
    for (int v = 0; v < 8; v++) d[v] = aeidx[N_EDGES + base + v + 8 * half];
#pragma unroll
    for (int n = 0; n < 8; n++)
#pragma unroll
        for (int v = 0; v < 8; v++) {
            size_t idx = (size_t)(base + v + 8 * half) * DM + (n << 4) + l15;
            atomicAdd(&atomacc[(size_t)d[v] * DM + (n << 4) + l15],
                      acc[n][v] * mji[idx]);
        }
}

// atom = swish(atom@dW0+db0); atom = swish(atom@dW1+db1); out2 = atom@oW (128->2)
__global__ __launch_bounds__(256, 1)
void k_atoms(const float* __restrict__ aacc, const float* __restrict__ dW,
             const float* __restrict__ db, const float* __restrict__ oW,
             float* __restrict__ out2) {
    __shared__ float sG[8 * 2048];
    int wave = threadIdx.x >> 5, lane = threadIdx.x & 31;
    int tile = blockIdx.x * 8 + wave;
    if (tile >= (N_ATOMS / 16)) return;   // wave-level guard, no barriers in kernel
    int half = lane >> 4, l15 = lane & 15;
    int base = tile * 16;

    v8f acc[8] = {};
    for (int kk = 0; kk < 32; kk++) {
        v2f a = ldAg(aacc, base, DM, kk * 4, lane);
#pragma unroll
        for (int n = 0; n < 8; n++) acc[n] = wmma4(a, ldBg(dW, kk * 4, n, lane), acc[n]);
    }
#pragma unroll
    for (int n = 0; n < 8; n++) {
        float bb = db[(n << 4) + l15];
#pragma unroll
        for (int v = 0; v < 8; v++)
            sG[wave * 2048 + (v + 8 * half) * 128 + (n << 4) + l15] =
                swishf(acc[n][v] + bb);
    }
    v8f acc2[8] = {};
    for (int kk = 0; kk < 32; kk++) {
        int k = kk * 4 + half * 2;
        const float* ap = sG + wave * 2048 + l15 * 128 + k;
        v2f a; a.x = ap[0]; a.y = ap[1];
#pragma unroll
        for (int n = 0; n < 8; n++)
            acc2[n] = wmma4(a, ldBg(dW + 16384, kk * 4, n, lane), acc2[n]);
    }
#pragma unroll
    for (int n = 0; n < 8; n++) {
        float bb = db[128 + (n << 4) + l15];
#pragma unroll
        for (int v = 0; v < 8; v++)
            sG[wave * 2048 + (v + 8 * half) * 128 + (n << 4) + l15] =
                swishf(acc2[n][v] + bb);
    }
    if (lane < 16) {
        const float* hr = sG + wave * 2048 + lane * 128;
        float o0 = 0.0f, o1 = 0.0f;
        for (int k = 0; k < 128; k++) {
            float h = hr[k];
            o0 += h * oW[k * 2];
            o1 += h * oW[k * 2 + 1];
        }
        out2[(size_t)(base + lane) * 2]     = o0;
        out2[(size_t)(base + lane) * 2 + 1] = o1;
    }
}

__global__ void k_reg(float* __restrict__ p) {
    if (threadIdx.x == 0 && blockIdx.x == 0) *p = 0.0f;
}

extern "C" void kernel_launch(void* const* d_in, const int* in_sizes, int n_in,
                              void* d_out, int out_size, void* d_ws, size_t ws_size,
                              hipStream_t stream) {
    const float* msg       = (const float*)d_in[0];
    const float* rbf       = (const float*)d_in[1];
    const float* sbf       = (const float*)d_in[2];
    const float* gate      = (const float*)d_in[3];
    const float* W_rbf     = (const float*)d_in[4];
    const float* lin_msg_W = (const float*)d_in[5];
    const float* lin_msg_b = (const float*)d_in[6];
    const float* W_sbf     = (const float*)d_in[7];
    const float* W_bil     = (const float*)d_in[8];
    const float* riW1 = (const float*)d_in[9];
    const float* rib1 = (const float*)d_in[10];
    const float* riW2 = (const float*)d_in[11];
    const float* rib2 = (const float*)d_in[12];
    const float* liW  = (const float*)d_in[13];
    const float* lib  = (const float*)d_in[14];
    const float* rmW1 = (const float*)d_in[15];
    const float* rmb1 = (const float*)d_in[16];
    const float* rmW2 = (const float*)d_in[17];
    const float* rmb2 = (const float*)d_in[18];
    const float* oWr  = (const float*)d_in[19];
    const float* dW   = (const float*)d_in[20];
    const float* db   = (const float*)d_in[21];
    const float* oW   = (const float*)d_in[22];
    const int* meidx  = (const int*)d_in[23];
    const int* aeidx  = (const int*)d_in[24];

    float* mji_out = (float*)d_out;
    float* out2    = mji_out + (size_t)N_EDGES * DM;
    float* reg     = out2 + (size_t)N_ATOMS * 2;

    float* xbuf    = (float*)d_ws;
    float* accbuf  = xbuf + (size_t)N_EDGES * DM;
    float* atombuf = accbuf + (size_t)N_EDGES * DM;

    const int eblocks = N_EDGES / 16 / 8;       // 3125
    const int mblocks = N_MSG / 16 / 8;         // 9375
    const int ablocks = (N_ATOMS / 16 + 7) / 8; // 313

    // zero accumulation buffers (acc | atom are contiguous in ws)
    k_zero<<<2048, 256, 0, stream>>>(accbuf,
        (size_t)N_EDGES * DM + (size_t)N_ATOMS * DM);

    // edge features
    k_edge_x<<<eblocks, 256, 0, stream>>>(msg, rbf, lin_msg_W, lin_msg_b, W_rbf, xbuf);

    // bilinear line-graph message passing + scatter (132 KB dynamic LDS)
    k_bilinear<<<mblocks, 256, (16384 + 16384 + 336) * sizeof(float), stream>>>(
        xbuf, sbf, W_sbf, W_bil, meidx, accbuf);

    // residual interaction (acc -> xbuf, xbuf reused as scratch)
    k_resblock<<<eblocks, 256, 0, stream>>>(accbuf, xbuf, riW1, rib1, riW2, rib2);

    // interacted message + gated residual -> final mji buffer
    k_inter<<<eblocks, 256, 0, stream>>>(xbuf, msg, gate, liW, lib, mji_out);

    // two message residual blocks, in place (each wave owns its 16 rows)
    k_resblock<<<eblocks, 256, 0, stream>>>(mji_out, mji_out, rmW1, rmb1, rmW2, rmb2);
    k_resblock<<<eblocks, 256, 0, stream>>>(mji_out, mji_out,
        rmW1 + 16384, rmb1 + 128, rmW2 + 16384, rmb2 + 128);

    // rbf-gated scatter to atoms
    k_outedge<<<eblocks, 256, 0, stream>>>(rbf, oWr, mji_out, aeidx, atombuf);

    // dense stack + readout
    k_atoms<<<ablocks, 256, 0, stream>>>(atombuf, dW, db, oW, out2);

    // regularization scalar
    k_reg<<<1, 1, 0, stream>>>(reg);
}